// PP_Post_Bev_w_Head_3616362463428
// MI455X (gfx1250) — compile-verified
//
#include <hip/hip_runtime.h>
#include <hip/hip_bf16.h>

// ---------------- problem constants ----------------
#define HH 512
#define WW 512
#define NUM_CLS 3
#define NUM_ANCH 6
#define CODE 7
#define NTOT (HH * WW * NUM_ANCH)   // 1,572,864
#define PRE_K 4096
#define POST_K 500
#define SCORE_THRESH 0.1f
#define IOU_THRESH 0.01f
#define DIR_OFF 0.78539f
#define NBINS 1024
#define MASK_W (PRE_K / 32)         // 128 words per row
#define PI_F 3.14159265358979323846f

typedef __attribute__((ext_vector_type(2))) float v2f;
typedef __attribute__((ext_vector_type(8))) float v8f;

// ---------------- K0: zero histogram + counters ----------------
__global__ void k_init(unsigned* __restrict__ hist, unsigned* __restrict__ cnt) {
    int t = blockIdx.x * blockDim.x + threadIdx.x;
    if (t < NBINS) hist[t] = 0u;
    if (t < 16)    cnt[t]  = 0u;
}

// ---------------- K1: per-anchor score (sigmoid of max logit) + histogram ----------------
__global__ void k_scores(const float* __restrict__ cls,
                         float* __restrict__ scores,
                         unsigned* __restrict__ hist) {
    int n = blockIdx.x * blockDim.x + threadIdx.x;
    if (n >= NTOT) return;
    // stream-prefetch ahead (speculative; OOB silently dropped) -> global_prefetch_b8
    __builtin_prefetch(&cls[n * 3 + 4096], 0, 0);
    float c0 = cls[n * 3 + 0], c1 = cls[n * 3 + 1], c2 = cls[n * 3 + 2];
    float m = fmaxf(c0, fmaxf(c1, c2));
    float p = 1.0f / (1.0f + __expf(-m));      // max(sigmoid) == sigmoid(max)
    if (p < SCORE_THRESH) p = 0.0f;
    scores[n] = p;
    int b = (int)(p * (float)NBINS);
    b = b > (NBINS - 1) ? (NBINS - 1) : b;
    atomicAdd(&hist[b], 1u);
}

// ---------------- K2: find threshold bin for top-PRE_K; zero selection buffers --------
__global__ void k_thresh(const unsigned* __restrict__ hist, unsigned* __restrict__ cnt,
                         float* __restrict__ sel_s, int* __restrict__ sel_i) {
    int t = threadIdx.x;
    for (int k = t; k < PRE_K; k += blockDim.x) { sel_s[k] = 0.0f; sel_i[k] = 0; }
    if (t == 0) {
        unsigned suffix = 0, above = 0;
        int T = -1;
        for (int b = NBINS - 1; b >= 0; --b) {
            unsigned ns = suffix + hist[b];
            if (ns >= PRE_K) { T = b; above = suffix; break; }
            suffix = ns;
        }
        if (T < 0) { T = 0; above = suffix - hist[0]; }
        cnt[2] = (unsigned)T;
        cnt[3] = above;
    }
}

// ---------------- K3: compact candidates above / at threshold bin ----------------
__global__ void k_compact(const float* __restrict__ scores, unsigned* __restrict__ cnt,
                          float* __restrict__ sel_s, int* __restrict__ sel_i) {
    int n = blockIdx.x * blockDim.x + threadIdx.x;
    if (n >= NTOT) return;
    float p = scores[n];
    int b = (int)(p * (float)NBINS);
    b = b > (NBINS - 1) ? (NBINS - 1) : b;
    int T = (int)cnt[2];
    unsigned above = cnt[3];
    if (b > T) {
        unsigned pos = atomicAdd(&cnt[0], 1u);
        if (pos < PRE_K) { sel_s[pos] = p; sel_i[pos] = n; }
    } else if (b == T) {
        unsigned pos = above + atomicAdd(&cnt[1], 1u);
        if (pos < PRE_K) { sel_s[pos] = p; sel_i[pos] = n; }
    }
}

// ---------------- K4: single-block bitonic sort (score desc, idx asc) ----------------
__global__ void k_sort(float* __restrict__ sel_s, int* __restrict__ sel_i) {
    __shared__ float ss[PRE_K];
    __shared__ int   si[PRE_K];
    int t = threadIdx.x;  // 1024 threads
    for (int e = 0; e < 4; ++e) {
        ss[t + e * 1024] = sel_s[t + e * 1024];
        si[t + e * 1024] = sel_i[t + e * 1024];
    }
    __syncthreads();
    for (int k = 2; k <= PRE_K; k <<= 1) {
        for (int j = k >> 1; j > 0; j >>= 1) {
            for (int e = 0; e < 4; ++e) {
                int i = t + e * 1024;
                int ixj = i ^ j;
                if (ixj > i) {
                    float a = ss[i], bm = ss[ixj];
                    int   ai = si[i], bi = si[ixj];
                    bool desc = ((i & k) == 0);
                    bool r_ji = (bm > a) || (bm == a && bi < ai);
                    bool r_ij = (a > bm) || (a == bm && ai < bi);
                    bool sw = desc ? r_ji : r_ij;
                    if (sw) { ss[i] = bm; ss[ixj] = a; si[i] = bi; si[ixj] = ai; }
                }
            }
            __syncthreads();
        }
    }
    for (int e = 0; e < 4; ++e) {
        sel_s[t + e * 1024] = ss[t + e * 1024];
        sel_i[t + e * 1024] = si[t + e * 1024];
    }
}

// ---------------- K5: gather-decode only the selected 4096 boxes ----------------
__global__ void k_decode(const int* __restrict__ ti, const float* __restrict__ box,
                         const float* __restrict__ anch, const float* __restrict__ dirp,
                         const float* __restrict__ cls,
                         float* __restrict__ tb, int* __restrict__ tl,
                         float* __restrict__ bx1, float* __restrict__ bx2,
                         float* __restrict__ by1, float* __restrict__ by2,
                         float* __restrict__ barea) {
    int k = blockIdx.x * blockDim.x + threadIdx.x;
    if (k >= PRE_K) return;
    int n = ti[k];
    const float* a  = anch + (size_t)n * CODE;
    const float* bx = box  + (size_t)n * CODE;
    float xa = a[0], ya = a[1], za = a[2], dxa = a[3], dya = a[4], dza = a[5], ra = a[6];
    float xt = bx[0], yt = bx[1], zt = bx[2], dxt = bx[3], dyt = bx[4], dzt = bx[5], rt = bx[6];
    za += dza * 0.5f;
    float diag = sqrtf(dxa * dxa + dya * dya);
    float xg  = xt * diag + xa;
    float yg  = yt * diag + ya;
    float zg  = zt * dza + za;
    float dxg = __expf(dxt) * dxa;
    float dyg = __expf(dyt) * dya;
    float dzg = __expf(dzt) * dza;
    float rg  = rt + ra;
    zg -= dzg * 0.5f;
    float d0 = dirp[n * 2 + 0], d1 = dirp[n * 2 + 1];
    float dirl = (d1 > d0) ? 1.0f : 0.0f;          // argmax tie -> index 0
    float val = rg - DIR_OFF;
    val = val - floorf(val / PI_F) * PI_F;
    float rot = val + DIR_OFF + PI_F * dirl;
    float* o = tb + (size_t)k * CODE;
    o[0] = xg; o[1] = yg; o[2] = zg; o[3] = dxg; o[4] = dyg; o[5] = dzg; o[6] = rot;
    float c0 = cls[n * 3 + 0], c1 = cls[n * 3 + 1], c2 = cls[n * 3 + 2];
    int lab = 0; float best = c0;
    if (c1 > best) { best = c1; lab = 1; }
    if (c2 > best) { lab = 2; }
    tl[k] = lab + 1;
    float x1 = xg - dxg * 0.5f, x2 = xg + dxg * 0.5f;
    float y1 = yg - dyg * 0.5f, y2 = yg + dyg * 0.5f;
    bx1[k] = x1; bx2[k] = x2; by1[k] = y1; by2[k] = y2;
    barea[k] = (x2 - x1) * (y2 - y1);
}

// ---------------- K6: suppression bitmask; area[i]+area[j] via rank-2 f32 WMMA -------
// grid: (MASK_W, PRE_K/16), block: 64 threads (2 waves); each wave does a 16x16 subtile.
// Ballot results are wave-uniform: transpose them into per-lane row words with a
// cndmask select chain; each wave plain-stores its disjoint LDS half (no atomics).
__global__ void k_iou_mask(const float* __restrict__ bx1, const float* __restrict__ bx2,
                           const float* __restrict__ by1, const float* __restrict__ by2,
                           const float* __restrict__ barea, unsigned* __restrict__ gmask) {
    __shared__ float rx1[16], rx2[16], ry1[16], ry2[16], rar[16];
    __shared__ float cx1[32], cx2[32], cy1[32], cy2[32], car[32];
    __shared__ unsigned smw[2][16];     // [wave][row]: 16-bit half-row masks
    int wj = blockIdx.x;          // word column 0..127
    int ri = blockIdx.y;          // row tile 0..255
    int tid = threadIdx.x;
    int i0 = ri * 16, j0 = wj * 32;
    if (tid < 16) {
        rx1[tid] = bx1[i0 + tid]; rx2[tid] = bx2[i0 + tid];
        ry1[tid] = by1[i0 + tid]; ry2[tid] = by2[i0 + tid];
        rar[tid] = barea[i0 + tid];
    }
    if (tid < 32) {
        cx1[tid] = bx1[j0 + tid]; cx2[tid] = bx2[j0 + tid];
        cy1[tid] = by1[j0 + tid]; cy2[tid] = by2[j0 + tid];
        car[tid] = barea[j0 + tid];
    }
    __syncthreads();
    int wave = tid >> 5, lane = tid & 31;
    int cb = wave * 16;           // column sub-base within the 32-wide tile
    int l15 = lane & 15;
    // A (16x4): row m -> [area_row[m], 1, 0, 0]; lanes>=16 hold K=2,3 (zero).
    // B (4x16): rows   -> [1; area_col; 0; 0];    lanes>=16 hold K=2,3 (zero).
    float sel = (lane < 16) ? 1.0f : 0.0f;       // branch-free: EXEC stays all-1s for WMMA
    v2f a, b;
    a.x = rar[l15] * sel;       a.y = sel;
    b.x = sel;                  b.y = car[cb + l15] * sel;
    v8f acc = {};
    // D[m][n] = area_row[m] + area_col[n]
    acc = __builtin_amdgcn_wmma_f32_16x16x4_f32(false, a, false, b, (short)0, acc,
                                                false, false);
    unsigned ball[8];
#pragma unroll
    for (int v = 0; v < 8; ++v) {
        int m = v + ((lane & 16) ? 8 : 0);       // C/D layout: lanes 16-31 hold M = v+8
        int n = lane & 15;
        float ix = fminf(rx2[m], cx2[cb + n]) - fmaxf(rx1[m], cx1[cb + n]);
        float iy = fminf(ry2[m], cy2[cb + n]) - fmaxf(ry1[m], cy1[cb + n]);
        ix = fmaxf(ix, 0.0f); iy = fmaxf(iy, 0.0f);
        float inter = ix * iy;
        // iou > t  <=>  inter > t*(areasum - inter + 1e-6): denom > 0 always
        // (inter <= min(area_i, area_j); exp-decoded dims => areas > 0)
        bool pred = inter > IOU_THRESH * (acc[v] - inter + 1e-6f);
        ball[v] = (unsigned)(__ballot(pred) & 0xFFFFFFFFull);
    }
    // transpose uniform ballots -> per-lane row word (lane l gets bits of row l)
    unsigned myrow = 0;
#pragma unroll
    for (int v = 0; v < 8; ++v) {
        myrow = (l15 == v)     ? (ball[v] & 0xFFFFu) : myrow;   // rows 0..7: low half
        myrow = (l15 == v + 8) ? (ball[v] >> 16)     : myrow;   // rows 8..15: high half
    }
    if (lane < 16) smw[wave][lane] = myrow;      // disjoint per wave: plain ds_store
    __syncthreads();
    if (tid < 16)
        gmask[(size_t)(i0 + tid) * MASK_W + wj] = smw[0][tid] | (smw[1][tid] << 16);
}

// ---------------- K7: greedy NMS scan, remv formulation (single wave32) ----------------
// Semantically identical to the reference scan: i kept iff valid and no kept j<i has
// iou>thresh. Mask rows are loaded ONLY for kept rows; suppressed rows cost one
// wave-uniform LDS bit test. Full-row OR is safe: bits <= i are never consulted again.
__global__ void k_nms_scan(const unsigned* __restrict__ gmask, const float* __restrict__ ts,
                           unsigned* __restrict__ gkeep) {
    __shared__ unsigned remv[MASK_W];
    __shared__ unsigned keepw[MASK_W];
    int lane = threadIdx.x;  // 32 threads
    for (int e = 0; e < 4; ++e) { remv[lane + e * 32] = 0u; keepw[lane + e * 32] = 0u; }
    __syncthreads();
    for (int i = 0; i < PRE_K; ++i) {
        int iw = i >> 5, ib = i & 31;
        bool supp = (remv[iw] >> ib) & 1u;          // uniform LDS broadcast read
        bool ki = (ts[i] > 0.0f) && !supp;          // wave-uniform -> whole-wave branch
        if (ki) {
            if (lane == 0) keepw[iw] |= 1u << ib;
#pragma unroll
            for (int e = 0; e < 4; ++e) {
                int w = lane + e * 32;               // each lane owns 4 disjoint words
                remv[w] |= gmask[(size_t)i * MASK_W + w];
            }
        }
        __syncthreads();
    }
    for (int e = 0; e < 4; ++e) gkeep[lane + e * 32] = keepw[lane + e * 32];
}

// ---------------- K8: final top-POST_K selection + output assembly ----------------
__global__ void k_final(const unsigned* __restrict__ gkeep, const float* __restrict__ ts,
                        const float* __restrict__ tb, const int* __restrict__ tl,
                        float* __restrict__ out) {
    __shared__ int   fi[POST_K];
    __shared__ float fsv[POST_K];
    int t = threadIdx.x;  // 512 threads
    if (t == 0) {
        // ts is sorted descending: kept entries (in order) are the largest final
        // scores, then zeros tie-broken by ascending index — matches top_k.
        int cnt = 0;
        for (int i = 0; i < PRE_K && cnt < POST_K; ++i)
            if (gkeep[i >> 5] & (1u << (i & 31))) { fi[cnt] = i; fsv[cnt] = ts[i]; ++cnt; }
        for (int i = 0; i < PRE_K && cnt < POST_K; ++i)
            if (!(gkeep[i >> 5] & (1u << (i & 31)))) { fi[cnt] = i; fsv[cnt] = 0.0f; ++cnt; }
    }
    __syncthreads();
    if (t < POST_K) {
        int k = fi[t];
        for (int c = 0; c < CODE; ++c) out[t * 8 + c] = tb[k * CODE + c];
        out[t * 8 + 7] = fsv[t];
        out[POST_K * 8 + t] = (float)tl[k];   // labels written as float into d_out tail
    }
}

// ---------------- host-side orchestration ----------------
extern "C" void kernel_launch(void* const* d_in, const int* in_sizes, int n_in,
                              void* d_out, int out_size, void* d_ws, size_t ws_size,
                              hipStream_t stream) {
    const float* cls  = (const float*)d_in[0];   // (1,512,512,18) -> flat (N,3)
    const float* box  = (const float*)d_in[1];   // (1,512,512,42) -> flat (N,7)
    const float* dirp = (const float*)d_in[2];   // (1,512,512,12) -> flat (N,2)
    const float* anch = (const float*)d_in[3];   // (N,7)
    float* out = (float*)d_out;                  // 500*8 boxes+score, then 500 labels

    // workspace layout (~8.7 MB total)
    char* base = (char*)d_ws;
    float*    scores = (float*)base;                           // N
    unsigned* hist   = (unsigned*)(base + (size_t)NTOT * 4);   // 1024
    unsigned* cnt    = hist + NBINS;                           // 16
    float*    sel_s  = (float*)(cnt + 16);                     // PRE_K
    int*      sel_i  = (int*)(sel_s + PRE_K);                  // PRE_K
    float*    tb     = (float*)(sel_i + PRE_K);                // PRE_K*7
    int*      tl     = (int*)(tb + (size_t)PRE_K * CODE);      // PRE_K
    float*    bx1    = (float*)(tl + PRE_K);
    float*    bx2    = bx1 + PRE_K;
    float*    by1    = bx2 + PRE_K;
    float*    by2    = by1 + PRE_K;
    float*    barea  = by2 + PRE_K;
    unsigned* gmask  = (unsigned*)(barea + PRE_K);             // PRE_K*128
    unsigned* gkeep  = gmask + (size_t)PRE_K * MASK_W;         // 128

    k_init   <<<dim3(4), 256, 0, stream>>>(hist, cnt);
    k_scores <<<dim3(NTOT / 256), 256, 0, stream>>>(cls, scores, hist);
    k_thresh <<<dim3(1), 1024, 0, stream>>>(hist, cnt, sel_s, sel_i);
    k_compact<<<dim3(NTOT / 256), 256, 0, stream>>>(scores, cnt, sel_s, sel_i);
    k_sort   <<<dim3(1), 1024, 0, stream>>>(sel_s, sel_i);
    k_decode <<<dim3(PRE_K / 256), 256, 0, stream>>>(sel_i, box, anch, dirp, cls,
                                                     tb, tl, bx1, bx2, by1, by2, barea);
    k_iou_mask<<<dim3(MASK_W, PRE_K / 16), 64, 0, stream>>>(bx1, bx2, by1, by2, barea, gmask);
    k_nms_scan<<<dim3(1), 32, 0, stream>>>(gmask, sel_s, gkeep);
    k_final  <<<dim3(1), 512, 0, stream>>>(gkeep, sel_s, tb, tl, out);
}